// LocalAttention_34462817583607
// MI455X (gfx1250) — compile-verified
//
#include <hip/hip_runtime.h>

typedef __attribute__((ext_vector_type(2))) float v2f;
typedef __attribute__((ext_vector_type(8))) float v8f;

#define B_   8
#define C_   64
#define O_   64
#define H_   128
#define W_   128
#define HW_  (H_ * W_)
#define KH_  7
#define KW_  7
#define PAD_ 3
#define KK_  (KH_ * KW_)      // 49

// attention kernel tiling: 32x16 pixels per block, 2 px per thread (in w)
#define TW_   32
#define TH_   16
#define HALOH_ (TH_ + 2 * PAD_)   // 22
#define HALOW_ (TW_ + 2 * PAD_)   // 38
#define LROW_  (HALOW_ + 1)       // 39 (odd -> bank-parity split for stride-2)
#define CCH_  8                   // channel chunk staged in LDS

// ---------------------------------------------------------------------------
// Kernel 1: q/k/v = W{1,2,3} @ x  (1x1 convs as GEMM) via V_WMMA_F32_16X16X4_F32
// One wave32 handles 16 consecutive pixels (one row segment) x all 64 out chans
// for all three weight matrices. K-loop fully unrolled: 16 steps x 12 WMMAs.
//
// Fragment layouts (ISA 7.12.2, wave32):
//   A (16x4, 2 VGPR): lane m (m=lane&15, half=lane>>4): v.x = A[m, 2*half],
//                                                       v.y = A[m, 2*half+1]
//   B (4x16, 2 VGPR): lane n: v.x = B[2*half, n], v.y = B[2*half+1, n]
//   C/D (16x16, 8 VGPR): comp r: lanes0-15 -> (M=r, N=lane),
//                                lanes16-31 -> (M=r+8, N=lane-16)
// ---------------------------------------------------------------------------
__global__ __launch_bounds__(256, 1)
void qkv_wmma(const float* __restrict__ x,
              const float* __restrict__ W1,
              const float* __restrict__ W2,
              const float* __restrict__ W3,
              float* __restrict__ q,
              float* __restrict__ k,
              float* __restrict__ v)
{
    const int lane = threadIdx.x & 31;
    const int wave = threadIdx.x >> 5;
    const int m    = lane & 15;
    const int half = lane >> 4;

    const long pixblk = (long)blockIdx.x * 8 + wave; // 16 pixels per wave
    const long p0     = pixblk * 16;                 // flattened pixel base
    const int  b      = (int)(p0 / HW_);
    const int  pin    = (int)(p0 % HW_);             // W=128 % 16 == 0 -> same row

    const float* xb = x + (long)b * C_ * HW_ + pin;

    v8f acc[3][4];
#pragma unroll
    for (int i = 0; i < 3; ++i)
#pragma unroll
        for (int j = 0; j < 4; ++j) {
            v8f z = {0.f, 0.f, 0.f, 0.f, 0.f, 0.f, 0.f, 0.f};
            acc[i][j] = z;
        }

#pragma unroll
    for (int c0 = 0; c0 < C_; c0 += 4) {
        const int kc = c0 + 2 * half;
        v2f bf;
        bf.x = xb[(long)(kc)     * HW_ + m];
        bf.y = xb[(long)(kc + 1) * HW_ + m];

#pragma unroll
        for (int mt = 0; mt < 4; ++mt) {
            const int o = mt * 16 + m;
            v2f a;
            a.x = W1[o * C_ + kc];
            a.y = W1[o * C_ + kc + 1];
            acc[0][mt] = __builtin_amdgcn_wmma_f32_16x16x4_f32(
                false, a, false, bf, (short)0, acc[0][mt], false, false);
            a.x = W2[o * C_ + kc];
            a.y = W2[o * C_ + kc + 1];
            acc[1][mt] = __builtin_amdgcn_wmma_f32_16x16x4_f32(
                false, a, false, bf, (short)0, acc[1][mt], false, false);
            a.x = W3[o * C_ + kc];
            a.y = W3[o * C_ + kc + 1];
            acc[2][mt] = __builtin_amdgcn_wmma_f32_16x16x4_f32(
                false, a, false, bf, (short)0, acc[2][mt], false, false);
        }
    }

    float* outs[3] = {q, k, v};
#pragma unroll
    for (int mat = 0; mat < 3; ++mat) {
        float* dst = outs[mat] + (long)b * O_ * HW_ + pin + m;
#pragma unroll
        for (int mt = 0; mt < 4; ++mt)
#pragma unroll
            for (int r = 0; r < 8; ++r) {
                const int o = mt * 16 + half * 8 + r;
                dst[(long)o * HW_] = acc[mat][mt][r];
            }
    }
}

// ---------------------------------------------------------------------------
// Kernel 2: fused scores (7x7 channel dots) -> softmax -> weighted sum.
// 32x16 pixel tile per block; each thread owns 2 horizontally adjacent pixels
// so one 8-float LDS strip feeds 14 FMAs (7 taps x 2 px). k/v halos staged in
// LDS 8 channels at a time; both 49-tap softmaxes fully register-resident.
// ---------------------------------------------------------------------------
__global__ __launch_bounds__(256, 1)
void local_attn(const float* __restrict__ q,
                const float* __restrict__ k,
                const float* __restrict__ v,
                float* __restrict__ out)
{
    __shared__ float tile[CCH_][HALOH_][LROW_];

    const int tx  = threadIdx.x;        // 0..15
    const int ty  = threadIdx.y;        // 0..15
    const int tid = ty * 16 + tx;
    const int b   = blockIdx.z;
    const int h0  = blockIdx.y * TH_;
    const int w0  = blockIdx.x * TW_;
    const int h   = h0 + ty;
    const int wA  = w0 + 2 * tx;        // pixel 0
    const int wB  = wA + 1;             // pixel 1
    const int lx  = 2 * tx;             // local col of pixel 0 (halo idx = lx+dw)

    const long baseB = (long)b * C_ * HW_;
    const long rowA  = baseB + (long)h * W_ + wA;   // + c*HW_ later

    float s0[KK_], s1[KK_];
#pragma unroll
    for (int i = 0; i < KK_; ++i) { s0[i] = 0.f; s1[i] = 0.f; }

    // ---- scores: s[d] = sum_c q[c,p] * k[c, p+d]  (zero-padded halo) ----
    for (int c0 = 0; c0 < C_; c0 += CCH_) {
        __syncthreads();
        for (int i = tid; i < CCH_ * HALOH_ * HALOW_; i += 256) {
            const int c  = i / (HALOH_ * HALOW_);
            const int r  = i % (HALOH_ * HALOW_);
            const int hh = r / HALOW_;
            const int ww = r % HALOW_;
            const int gh = h0 + hh - PAD_;
            const int gw = w0 + ww - PAD_;
            float val = 0.f;
            if (gh >= 0 && gh < H_ && gw >= 0 && gw < W_)
                val = k[baseB + (long)(c0 + c) * HW_ + gh * W_ + gw];
            tile[c][hh][ww] = val;
        }
        __syncthreads();
#pragma unroll
        for (int c = 0; c < CCH_; ++c) {
            const float qc0 = q[rowA + (long)(c0 + c) * HW_];
            const float qc1 = q[rowA + (long)(c0 + c) * HW_ + 1];
#pragma unroll
            for (int dh = 0; dh < KH_; ++dh) {
                float t[KW_ + 1];
#pragma unroll
                for (int j = 0; j < KW_ + 1; ++j)
                    t[j] = tile[c][ty + dh][lx + j];
#pragma unroll
                for (int dw = 0; dw < KW_; ++dw) {
                    s0[dh * KW_ + dw] = fmaf(qc0, t[dw],     s0[dh * KW_ + dw]);
                    s1[dh * KW_ + dw] = fmaf(qc1, t[dw + 1], s1[dh * KW_ + dw]);
                }
            }
        }
    }

    // ---- softmax over the 49 taps (registers only), both pixels ----
    float mx0 = s0[0], mx1 = s1[0];
#pragma unroll
    for (int i = 1; i < KK_; ++i) { mx0 = fmaxf(mx0, s0[i]); mx1 = fmaxf(mx1, s1[i]); }
    float sum0 = 0.f, sum1 = 0.f;
#pragma unroll
    for (int i = 0; i < KK_; ++i) {
        s0[i] = __expf(s0[i] - mx0); sum0 += s0[i];
        s1[i] = __expf(s1[i] - mx1); sum1 += s1[i];
    }
    const float inv0 = 1.f / sum0, inv1 = 1.f / sum1;
#pragma unroll
    for (int i = 0; i < KK_; ++i) { s0[i] *= inv0; s1[i] *= inv1; }

    // ---- weighting: out[o,p] = sum_d s[d] * v[o, p+d] ----
    for (int o0 = 0; o0 < O_; o0 += CCH_) {
        __syncthreads();
        for (int i = tid; i < CCH_ * HALOH_ * HALOW_; i += 256) {
            const int c  = i / (HALOH_ * HALOW_);
            const int r  = i % (HALOH_ * HALOW_);
            const int hh = r / HALOW_;
            const int ww = r % HALOW_;
            const int gh = h0 + hh - PAD_;
            const int gw = w0 + ww - PAD_;
            float val = 0.f;
            if (gh >= 0 && gh < H_ && gw >= 0 && gw < W_)
                val = v[baseB + (long)(o0 + c) * HW_ + gh * W_ + gw];
            tile[c][hh][ww] = val;
        }
        __syncthreads();
#pragma unroll
        for (int o = 0; o < CCH_; ++o) {
            float acc0 = 0.f, acc1 = 0.f;
#pragma unroll
            for (int dh = 0; dh < KH_; ++dh) {
                float t[KW_ + 1];
#pragma unroll
                for (int j = 0; j < KW_ + 1; ++j)
                    t[j] = tile[o][ty + dh][lx + j];
#pragma unroll
                for (int dw = 0; dw < KW_; ++dw) {
                    acc0 = fmaf(s0[dh * KW_ + dw], t[dw],     acc0);
                    acc1 = fmaf(s1[dh * KW_ + dw], t[dw + 1], acc1);
                }
            }
            out[rowA + (long)(o0 + o) * HW_]     = acc0;
            out[rowA + (long)(o0 + o) * HW_ + 1] = acc1;
        }
    }
    (void)wB;
}

// ---------------------------------------------------------------------------
extern "C" void kernel_launch(void* const* d_in, const int* in_sizes, int n_in,
                              void* d_out, int out_size, void* d_ws, size_t ws_size,
                              hipStream_t stream)
{
    const float* x  = (const float*)d_in[0];
    const float* W1 = (const float*)d_in[1];
    const float* W2 = (const float*)d_in[2];
    const float* W3 = (const float*)d_in[3];
    // d_in[4]/d_in[5] = kH/kW; fixed to 7 at compile time (required for the
    // fully-register-resident 49-tap softmax).
    (void)in_sizes; (void)n_in; (void)out_size; (void)ws_size;

    const size_t N = (size_t)B_ * O_ * HW_;   // 8,388,608 elems per tensor
    float* qbuf = (float*)d_ws;
    float* kbuf = qbuf + N;
    float* vbuf = kbuf + N;
    float* out  = (float*)d_out;

    // Kernel 1: 131072 pixels / 16 per wave / 8 waves per block = 1024 blocks
    qkv_wmma<<<1024, 256, 0, stream>>>(x, W1, W2, W3, qbuf, kbuf, vbuf);

    // Kernel 2: one 32x16 pixel tile per block (2 px per thread in w)
    dim3 grid2(W_ / TW_, H_ / TH_, B_);
    dim3 blk2(16, 16, 1);
    local_attn<<<grid2, blk2, 0, stream>>>(qbuf, kbuf, vbuf, out);
}